// SpatialTransformer_87917980549181
// MI455X (gfx1250) — compile-verified
//
#include <hip/hip_runtime.h>
#include <hip/hip_bf16.h>

// 3D spatial transformer (trilinear grid_sample, align_corners=False, border pad)
// B=2, C=2, D=H=W=128. Memory-bound gather kernel, MI455X-tuned:
//   - flow streamed once with non-temporal loads (TH=NT)
//   - output streamed once with non-temporal stores (TH=NT)
//   - image gathers regular-temporal: 32MB image stays resident in 192MB L2
//   - x-corner pairs fetched as single 8B loads (halves gather instructions);
//     border case handled by shifting the pair window, never reads OOB
//   - global_prefetch_b8 on next grid-stride iteration's flow
//   - wave32: 32 lanes x 4 voxels = one full W=128 row per wave

typedef float v4f __attribute__((ext_vector_type(4)));
typedef float v2f __attribute__((ext_vector_type(2)));
typedef v2f v2f_a4 __attribute__((aligned(4)));   // 8B vector, 4B alignment

namespace {
constexpr int B = 2, C = 2, D = 128, H = 128, W = 128;
constexpr int N    = D * H * W;        // voxels per (batch, plane) = 2^21
constexpr int QPB  = N / 4;            // quads per batch           = 2^19
constexpr int NQ   = B * QPB;          // total quads               = 2^20
constexpr int QPB_SHIFT = 19;          // QPB = 1<<19
}

__device__ __forceinline__ void unnorm_clamp(float base_idx, float f, int dim,
                                             float& t, int& i0, int& i1) {
  // Reference math, step for step:
  //   g  = linspace(-1,1,dim)[i] + 2*f/(dim-1)
  //   ix = ((g+1)*dim - 1)*0.5 ; clip to [0, dim-1]
  const float inv = 2.0f / (float)(dim - 1);
  float g  = (-1.0f + base_idx * inv) + f * inv;
  float ix = ((g + 1.0f) * (float)dim - 1.0f) * 0.5f;
  ix = fminf(fmaxf(ix, 0.0f), (float)(dim - 1));
  float i0f = floorf(ix);
  t  = ix - i0f;
  i0 = (int)i0f;
  i1 = min(i0 + 1, dim - 1);
}

__device__ __forceinline__ v2f load_pair(const float* base) {
  return *(const v2f_a4*)base;
}

__global__ __launch_bounds__(256) void
SpatialTransformer_87917980549181_kernel(const float* __restrict__ image,
                                         const float* __restrict__ flow,
                                         float* __restrict__ out) {
  const int nth = gridDim.x * blockDim.x;

  for (int q = blockIdx.x * blockDim.x + threadIdx.x; q < NQ; q += nth) {
    const int b  = q >> QPB_SHIFT;              // batch
    const int s  = (q - (b << QPB_SHIFT)) * 4;  // linear voxel index in [D,H,W]
    const int x  = s & (W - 1);                 // s..s+3 share (z,y); W=128
    const int y  = (s >> 7) & (H - 1);
    const int z  = s >> 14;

    // ---- stream flow (read-once): non-temporal b128 loads --------------------
    const float* fb = flow + (size_t)b * 3 * N + s;
    const v4f fx = __builtin_nontemporal_load((const v4f*)(fb));
    const v4f fy = __builtin_nontemporal_load((const v4f*)(fb + N));
    const v4f fz = __builtin_nontemporal_load((const v4f*)(fb + 2 * (size_t)N));

    // ---- prefetch next grid-stride iteration's flow (global_prefetch_b8) ----
    const int qn = q + nth;
    if (qn < NQ) {
      const int bn = qn >> QPB_SHIFT;
      const int sn = (qn - (bn << QPB_SHIFT)) * 4;
      const float* fn = flow + (size_t)bn * 3 * N + sn;
      __builtin_prefetch(fn,                 0, 3);
      __builtin_prefetch(fn + N,             0, 3);
      __builtin_prefetch(fn + 2 * (size_t)N, 0, 3);
    }

    const float* p0 = image + (size_t)(b * C) * N;  // channel 0 plane
    const float* p1 = p0 + N;                       // channel 1 plane

    v4f acc0, acc1;

#pragma unroll
    for (int e = 0; e < 4; ++e) {
      float tx, ty, tz;
      int x0, x1, y0, y1, z0, z1;
      unnorm_clamp((float)(x + e), fx[e], W, tx, x0, x1);
      unnorm_clamp((float)y,       fy[e], H, ty, y0, y1);
      unnorm_clamp((float)z,       fz[e], D, tz, z0, z1);
      (void)x1;

      // Pair-load window: xb = min(x0, W-2). Always in-bounds; at the border
      // (x0==W-1, x1 clamps to x0) both corners resolve to q.y = val[W-1].
      const bool xedge = (x0 == W - 1);
      const int  xb    = xedge ? (W - 2) : x0;

      // 4 row base addresses within a channel plane
      const int a00 = (z0 * H + y0) * W + xb;
      const int a01 = (z0 * H + y1) * W + xb;
      const int a10 = (z1 * H + y0) * W + xb;
      const int a11 = (z1 * H + y1) * W + xb;

      // 8 pair gathers (4 rows x 2 channels) instead of 16 scalar gathers
      const v2f q000 = load_pair(p0 + a00);
      const v2f q010 = load_pair(p0 + a01);
      const v2f q100 = load_pair(p0 + a10);
      const v2f q110 = load_pair(p0 + a11);
      const v2f r000 = load_pair(p1 + a00);
      const v2f r010 = load_pair(p1 + a01);
      const v2f r100 = load_pair(p1 + a10);
      const v2f r110 = load_pair(p1 + a11);

      // low corner select (border duplicates the high element)
      const float v000 = xedge ? q000.y : q000.x;
      const float v010 = xedge ? q010.y : q010.x;
      const float v100 = xedge ? q100.y : q100.x;
      const float v110 = xedge ? q110.y : q110.x;
      const float u000 = xedge ? r000.y : r000.x;
      const float u010 = xedge ? r010.y : r010.x;
      const float u100 = xedge ? r100.y : r100.x;
      const float u110 = xedge ? r110.y : r110.x;

      // 12-mul weight tree, shared across both channels
      const float wx0 = 1.0f - tx, wx1 = tx;
      const float wy0 = 1.0f - ty, wy1 = ty;
      const float wz0 = 1.0f - tz, wz1 = tz;
      const float w00 = wz0 * wy0, w01 = wz0 * wy1;
      const float w10 = wz1 * wy0, w11 = wz1 * wy1;
      const float w000 = w00 * wx0, w001 = w00 * wx1;
      const float w010 = w01 * wx0, w011 = w01 * wx1;
      const float w100 = w10 * wx0, w101 = w10 * wx1;
      const float w110 = w11 * wx0, w111 = w11 * wx1;

      // channel 0 blend
      float rr0;
      rr0 = v000 * w000;
      rr0 = fmaf(q000.y, w001, rr0);
      rr0 = fmaf(v010,   w010, rr0);
      rr0 = fmaf(q010.y, w011, rr0);
      rr0 = fmaf(v100,   w100, rr0);
      rr0 = fmaf(q100.y, w101, rr0);
      rr0 = fmaf(v110,   w110, rr0);
      rr0 = fmaf(q110.y, w111, rr0);
      acc0[e] = rr0;

      // channel 1 blend: same indices, same weights
      float rr1;
      rr1 = u000 * w000;
      rr1 = fmaf(r000.y, w001, rr1);
      rr1 = fmaf(u010,   w010, rr1);
      rr1 = fmaf(r010.y, w011, rr1);
      rr1 = fmaf(u100,   w100, rr1);
      rr1 = fmaf(r100.y, w101, rr1);
      rr1 = fmaf(u110,   w110, rr1);
      rr1 = fmaf(r110.y, w111, rr1);
      acc1[e] = rr1;
    }

    // ---- write-once output: non-temporal b128 stores -------------------------
    float* ob = out + (size_t)(b * C) * N + s;
    __builtin_nontemporal_store(acc0, (v4f*)ob);
    __builtin_nontemporal_store(acc1, (v4f*)(ob + N));
  }
}

extern "C" void kernel_launch(void* const* d_in, const int* in_sizes, int n_in,
                              void* d_out, int out_size, void* d_ws, size_t ws_size,
                              hipStream_t stream) {
  const float* image = (const float*)d_in[0];  // [B,C,D,H,W] f32
  const float* flow  = (const float*)d_in[1];  // [B,3,D,H,W] f32
  float* out = (float*)d_out;                  // [B,C,D,H,W] f32

  // 2048 blocks x 256 threads = 512K threads; NQ = 1M quads -> 2 grid-stride
  // iterations each (second iteration's flow is prefetched by the first).
  dim3 grid(2048), block(256);
  hipLaunchKernelGGL(SpatialTransformer_87917980549181_kernel, grid, block, 0,
                     stream, image, flow, out);
}